// DualPoolingGNN_86904368267866
// MI455X (gfx1250) — compile-verified
//
#include <hip/hip_runtime.h>
#include <hip/hip_bf16.h>
#include <math.h>

#define NN 50000
#define EE 800000
#define GG 512
#define DM 128
#define CC 10
#define LDA 132      // padded LDS row stride in floats (132 % 64 == 4 -> conflict-free column reads)
#define ROWS 80      // rows per GEMM block: 50000 = 80 * 625; 5 M-tiles per wave
#define MT 5         // M-tiles per wave

typedef __attribute__((ext_vector_type(2))) float v2f;
typedef __attribute__((ext_vector_type(8))) float v8f;
typedef __attribute__((ext_vector_type(4))) unsigned int u32x4;
typedef __attribute__((ext_vector_type(8))) int i32x8;
typedef __attribute__((ext_vector_type(4))) int i32x4;

// ---------------- h0 init: h0[i] = emb[x[i]]; also seeds the self-term sum buffer
__global__ void init_h0_kernel(const int* __restrict__ x, const float* __restrict__ emb,
                               float* __restrict__ h0, float* __restrict__ s) {
    int tid = blockIdx.x * blockDim.x + threadIdx.x;
    if (tid >= NN * DM) return;
    int row = tid >> 7, col = tid & 127;
    float v = emb[x[row] * DM + col];
    h0[tid] = v;
    s[tid] = v;
}

__global__ void copy_kernel(float* __restrict__ dst, const float* __restrict__ src, int n) {
    int tid = blockIdx.x * blockDim.x + threadIdx.x;
    if (tid < n) dst[tid] = src[tid];
}

// ---------------- edge scatter-add: one wave per edge, 4 floats per lane
__global__ void edge_scatter_kernel(const int* __restrict__ src, const int* __restrict__ dst,
                                    const float* __restrict__ h, float* __restrict__ s) {
    long long tid = (long long)blockIdx.x * blockDim.x + threadIdx.x;
    if (tid >= (long long)EE * 32) return;
    int e = (int)(tid >> 5);
    int l = (int)(tid & 31);
    int si = src[e], di = dst[e];
    const float4 v = *(const float4*)(h + (size_t)si * DM + l * 4);
    float* p = s + (size_t)di * DM + l * 4;
    atomicAdd(p + 0, v.x);
    atomicAdd(p + 1, v.y);
    atomicAdd(p + 2, v.z);
    atomicAdd(p + 3, v.w);
}

// ---------------- TDM: DMA an [rows x 128] f32 tile, padded to LDA-float rows in LDS
// D# per CDNA5 ISA ch.8: group0 = {count/flags, lds_addr, global_addr lo, global_addr hi|type},
// group1 = {dims/strides}. pad: 128 DWORDs interval (code 6), 4 DWORDs amount (code 3) -> row stride 132.
__device__ __forceinline__ void tdm_load_tile(const float* gsrc, unsigned lds_off, int rows) {
    unsigned long long ga = (unsigned long long)(uintptr_t)gsrc;
    u32x4 g0;
    g0[0] = 1u;                                         // count=1, user descriptor
    g0[1] = lds_off;                                    // lds_addr (bytes)
    g0[2] = (unsigned)(ga & 0xffffffffu);               // global_addr[31:0]
    g0[3] = (unsigned)((ga >> 32) & 0x01ffffffu)        // global_addr[56:32]
            | (2u << 30);                               // type = 2 ("image")
    i32x8 g1;
    g1[0] = (2 << 16)        // data_size = 4 bytes
          | (1 << 20)        // pad_enable
          | (6 << 22)        // pad_interval: 128 DWORDs
          | (3 << 25);       // pad_amount: 4 DWORDs
    g1[1] = (DM << 16);      // tensor_dim0[15:0] = 128 (bits 63:48)
    g1[2] = (rows << 16);    // tensor_dim0[31:16]=0 | tensor_dim1[15:0]=rows
    g1[3] = (DM << 16);      // tensor_dim1[31:16]=0 | tile_dim0 = 128
    g1[4] = rows & 0xffff;   // tile_dim1 = rows, tile_dim2 = 0
    g1[5] = DM;              // tensor_dim0_stride[31:0] = 128 elements
    g1[6] = 0;               // stride hi | tensor_dim1_stride lo
    g1[7] = 0;
    i32x4 gz4 = {0, 0, 0, 0};                 // groups 2/3: 2D tensor, all zero
    i32x8 gz8 = {0, 0, 0, 0, 0, 0, 0, 0};     // trailing group (6-arg toolchain form)
    __builtin_amdgcn_tensor_load_to_lds(g0, g1, gz4, gz4, gz8, 0);
}

// ---------------- shared WMMA compute: 8 waves x 5 M-tiles, K=128 -> 160 wmma/wave
__device__ __forceinline__ void wmma_compute_tiles(const float* __restrict__ atile,
                                                   const float* __restrict__ W,
                                                   const float* __restrict__ bias,
                                                   const float* __restrict__ residual,
                                                   float* __restrict__ out, int rowBase) {
    const int t    = threadIdx.x;
    const int w    = t >> 5;            // wave id -> 16-column B/D tile
    const int l    = t & 31;
    const int m    = l & 15;            // row (A) / column (B,D) within tile
    const int kh   = (l >> 4) << 1;     // half-wave K offset {0,2}
    const int ncol = (w << 4) + m;

    v8f acc[MT];
#pragma unroll
    for (int i = 0; i < MT; ++i) acc[i] = (v8f){};

#pragma unroll
    for (int k4 = 0; k4 < DM; k4 += 4) {
        v2f b;
        b.x = W[(size_t)(k4 + kh) * DM + ncol];
        b.y = W[(size_t)(k4 + kh + 1) * DM + ncol];
#pragma unroll
        for (int i = 0; i < MT; ++i) {
            v2f a;
            a.x = atile[(i * 16 + m) * LDA + k4 + kh];
            a.y = atile[(i * 16 + m) * LDA + k4 + kh + 1];
            acc[i] = __builtin_amdgcn_wmma_f32_16x16x4_f32(false, a, false, b,
                                                           (short)0, acc[i], false, false);
        }
    }

    const float bc = bias[ncol];
#pragma unroll
    for (int i = 0; i < MT; ++i) {
#pragma unroll
        for (int vv = 0; vv < 8; ++vv) {
            int row = rowBase + i * 16 + vv + ((l >> 4) << 3);
            float val = acc[i][vv] + bc;
            if (residual) val += residual[(size_t)row * DM + ncol];
            out[(size_t)row * DM + ncol] = val;
        }
    }
}

// ---------------- GEMM variant 1: plain (A @ W + bias), A tile staged via Tensor Data Mover
__global__ void wmma_gemm_tdm_kernel(const float* __restrict__ A, const float* __restrict__ W,
                                     const float* __restrict__ bias, float* __restrict__ out) {
    __shared__ float atile[ROWS * LDA];
    const int rowBase = blockIdx.x * ROWS;
    if (threadIdx.x < 32) {  // one wave issues the DMA; TENSORcnt tracks completion
        tdm_load_tile(A + (size_t)rowBase * DM, (unsigned)(uintptr_t)atile, ROWS);
        __builtin_amdgcn_s_wait_tensorcnt(0);
    }
    __syncthreads();
    wmma_compute_tiles(atile, W, bias, nullptr, out, rowBase);
}

// ---------------- GEMM variant 2: BN(scale,shift)+ReLU fused into LDS staging (+ optional residual)
__global__ void wmma_gemm_bn_kernel(const float* __restrict__ A, const float* __restrict__ W,
                                    const float* __restrict__ bias,
                                    const float* __restrict__ bn_scale,
                                    const float* __restrict__ bn_shift,
                                    const float* __restrict__ residual,
                                    float* __restrict__ out) {
    __shared__ float atile[ROWS * LDA];
    const int rowBase = blockIdx.x * ROWS;
    for (int idx = threadIdx.x; idx < ROWS * DM; idx += 256) {
        int r = idx >> 7, c = idx & 127;
        float v = A[(size_t)(rowBase + r) * DM + c];
        v = fmaf(v, bn_scale[c], bn_shift[c]);
        v = fmaxf(v, 0.0f);
        atile[r * LDA + c] = v;
    }
    __syncthreads();
    wmma_compute_tiles(atile, W, bias, residual, out, rowBase);
}

// ---------------- column statistics (sum, sumsq) for BN
__global__ void zero_stats_kernel(float* __restrict__ stats) {
    if (threadIdx.x < 256) stats[threadIdx.x] = 0.0f;
}

__global__ void col_stats_kernel(const float* __restrict__ z, float* __restrict__ stats) {
    const int c = threadIdx.x;  // 128 threads
    float s = 0.0f, ss = 0.0f;
    for (int r = blockIdx.x; r < NN; r += gridDim.x) {
        float v = z[(size_t)r * DM + c];
        s += v;
        ss += v * v;
    }
    atomicAdd(&stats[c], s);
    atomicAdd(&stats[DM + c], ss);
}

__global__ void finalize_bn_kernel(const float* __restrict__ stats,
                                   const float* __restrict__ g, const float* __restrict__ be,
                                   float* __restrict__ bn) {
    int c = threadIdx.x;  // 128 threads
    if (c >= DM) return;
    float mean = stats[c] / (float)NN;
    float var  = stats[DM + c] / (float)NN - mean * mean;
    float scale = g[c] * rsqrtf(var + 1e-5f);
    bn[c]      = scale;                  // scale
    bn[DM + c] = be[c] - mean * scale;   // shift
}

// ---------------- pooling
__global__ void pool_init_kernel(float* __restrict__ psum, float* __restrict__ pmax,
                                 float* __restrict__ pcnt) {
    int tid = blockIdx.x * blockDim.x + threadIdx.x;
    if (tid < GG * DM) {
        psum[tid] = 0.0f;
        pmax[tid] = -INFINITY;
    }
    if (tid < GG) pcnt[tid] = 0.0f;
}

__global__ void pool_atomic_kernel(const int* __restrict__ batch, const float* __restrict__ h,
                                   float* __restrict__ psum, float* __restrict__ pmax,
                                   float* __restrict__ pcnt) {
    long long tid = (long long)blockIdx.x * blockDim.x + threadIdx.x;
    if (tid >= (long long)NN * 32) return;
    int i = (int)(tid >> 5);
    int l = (int)(tid & 31);
    int g = batch[i];
    const float4 v = *(const float4*)(h + (size_t)i * DM + l * 4);
    float* ps = psum + (size_t)g * DM + l * 4;
    float* pm = pmax + (size_t)g * DM + l * 4;
    atomicAdd(ps + 0, v.x); atomicAdd(ps + 1, v.y);
    atomicAdd(ps + 2, v.z); atomicAdd(ps + 3, v.w);
    atomicMax(pm + 0, v.x); atomicMax(pm + 1, v.y);
    atomicMax(pm + 2, v.z); atomicMax(pm + 3, v.w);
    if (l == 0) atomicAdd(&pcnt[g], 1.0f);
}

__global__ void pool_finalize_kernel(const float* __restrict__ psum, const float* __restrict__ pmax,
                                     const float* __restrict__ pcnt,
                                     float* __restrict__ mean_repr, float* __restrict__ max_repr) {
    int tid = blockIdx.x * blockDim.x + threadIdx.x;
    if (tid >= GG * DM) return;
    int g = tid >> 7;
    float cnt = fmaxf(pcnt[g], 1.0f);
    mean_repr[tid] = psum[tid] / cnt;
    float mx = pmax[tid];
    max_repr[tid] = isfinite(mx) ? mx : 0.0f;
}

// ---------------- classification heads + ensemble (tiny: 512x10 dots of length 128)
__global__ void head_kernel(const float* __restrict__ mean_repr, const float* __restrict__ max_repr,
                            const float* __restrict__ Wm, const float* __restrict__ bm,
                            const float* __restrict__ Wx, const float* __restrict__ bx,
                            const float* __restrict__ ens_w,
                            float* __restrict__ ens_out, float* __restrict__ lm_out,
                            float* __restrict__ lx_out) {
    int tid = blockIdx.x * blockDim.x + threadIdx.x;
    if (tid >= GG * CC) return;
    int g = tid / CC, c = tid % CC;
    const float* mr = mean_repr + (size_t)g * DM;
    const float* xr = max_repr + (size_t)g * DM;
    float lm = bm[c], lx = bx[c];
#pragma unroll 4
    for (int d = 0; d < DM; ++d) {
        lm = fmaf(mr[d], Wm[d * CC + c], lm);
        lx = fmaf(xr[d], Wx[d * CC + c], lx);
    }
    float a0 = ens_w[0], a1 = ens_w[1];
    float mxe = fmaxf(a0, a1);
    float e0 = __expf(a0 - mxe), e1 = __expf(a1 - mxe);
    float w0 = e0 / (e0 + e1);
    ens_out[tid] = w0 * lm + (1.0f - w0) * lx;
    lm_out[tid] = lm;
    lx_out[tid] = lx;
}

extern "C" void kernel_launch(void* const* d_in, const int* in_sizes, int n_in,
                              void* d_out, int out_size, void* d_ws, size_t ws_size,
                              hipStream_t stream) {
    const int* x      = (const int*)d_in[0];
    const int* ei     = (const int*)d_in[1];
    const int* srcI   = ei;
    const int* dstI   = ei + EE;
    const int* batch  = (const int*)d_in[2];
    const float* emb  = (const float*)d_in[3];
    const float* W1a  = (const float*)d_in[4];
    const float* b1a  = (const float*)d_in[5];
    const float* g1   = (const float*)d_in[6];
    const float* be1  = (const float*)d_in[7];
    const float* W1b  = (const float*)d_in[8];
    const float* b1b  = (const float*)d_in[9];
    const float* W2a  = (const float*)d_in[10];
    const float* b2a  = (const float*)d_in[11];
    const float* g2   = (const float*)d_in[12];
    const float* be2  = (const float*)d_in[13];
    const float* W2b  = (const float*)d_in[14];
    const float* b2b  = (const float*)d_in[15];
    const float* Wm   = (const float*)d_in[16];
    const float* bm   = (const float*)d_in[17];
    const float* Wx   = (const float*)d_in[18];
    const float* bx   = (const float*)d_in[19];
    const float* ensw = (const float*)d_in[20];

    float* out = (float*)d_out;
    const size_t ND = (size_t)NN * DM;
    float* B0    = (float*)d_ws;          // h0 / h2
    float* B1    = B0 + ND;               // self + aggregate sum
    float* B2    = B1 + ND;               // z (pre-BN MLP hidden)
    float* B3    = B2 + ND;               // h1
    float* stats = B3 + ND;               // 256
    float* bn    = stats + 256;           // 256 (scale | shift)
    float* psum  = bn + 256;              // G*DM
    float* pmax  = psum + (size_t)GG * DM;
    float* pcnt  = pmax + (size_t)GG * DM;

    // output layout (flattened tuple, return order)
    float* ens_out   = out;                        // G*C
    float* lm_out    = out + GG * CC;              // G*C
    float* lx_out    = out + 2 * GG * CC;          // G*C
    float* mean_repr = out + 3 * GG * CC;          // G*DM
    float* max_repr  = mean_repr + GG * DM;        // G*DM

    const int TB = 256;
    const int gridND    = (int)((ND + TB - 1) / TB);
    const int gridEdge  = (int)(((long long)EE * 32 + TB - 1) / TB);
    const int gridGemm  = NN / ROWS;               // 50000 / 80 = 625
    const int gridPoolA = (int)(((long long)NN * 32 + TB - 1) / TB);
    const int gridGD    = (GG * DM + TB - 1) / TB;

    // ---------- layer 1
    init_h0_kernel<<<gridND, TB, 0, stream>>>(x, emb, B0, B1);
    edge_scatter_kernel<<<gridEdge, TB, 0, stream>>>(srcI, dstI, B0, B1);
    wmma_gemm_tdm_kernel<<<gridGemm, TB, 0, stream>>>(B1, W1a, b1a, B2);
    zero_stats_kernel<<<1, 256, 0, stream>>>(stats);
    col_stats_kernel<<<500, 128, 0, stream>>>(B2, stats);
    finalize_bn_kernel<<<1, 128, 0, stream>>>(stats, g1, be1, bn);
    wmma_gemm_bn_kernel<<<gridGemm, TB, 0, stream>>>(B2, W1b, b1b, bn, bn + DM, nullptr, B3);

    // ---------- layer 2 (residual +h1 fused into epilogue)
    copy_kernel<<<gridND, TB, 0, stream>>>(B1, B3, (int)ND);
    edge_scatter_kernel<<<gridEdge, TB, 0, stream>>>(srcI, dstI, B3, B1);
    wmma_gemm_tdm_kernel<<<gridGemm, TB, 0, stream>>>(B1, W2a, b2a, B2);
    zero_stats_kernel<<<1, 256, 0, stream>>>(stats);
    col_stats_kernel<<<500, 128, 0, stream>>>(B2, stats);
    finalize_bn_kernel<<<1, 128, 0, stream>>>(stats, g2, be2, bn);
    wmma_gemm_bn_kernel<<<gridGemm, TB, 0, stream>>>(B2, W2b, b2b, bn, bn + DM, B3, B0);

    // ---------- pooling + heads
    pool_init_kernel<<<gridGD, TB, 0, stream>>>(psum, pmax, pcnt);
    pool_atomic_kernel<<<gridPoolA, TB, 0, stream>>>(batch, B0, psum, pmax, pcnt);
    pool_finalize_kernel<<<gridGD, TB, 0, stream>>>(psum, pmax, pcnt, mean_repr, max_repr);
    head_kernel<<<(GG * CC + TB - 1) / TB, TB, 0, stream>>>(mean_repr, max_repr, Wm, bm, Wx, bx,
                                                            ensw, ens_out, lm_out, lx_out);
}